// PHOENIX_29119878267540
// MI455X (gfx1250) — compile-verified
//
#include <hip/hip_runtime.h>
#include <hip/hip_bf16.h>

#define N_USERS 100000
#define N_ITEMS 50000
#define N_NODES 150000
#define DD 64
#define NL 2
#define FF 2048
#define B_PAIRS 4096
#define E_TOTAL 1750000   // 1,600,000 random + 150,000 self-loops

typedef __attribute__((ext_vector_type(2))) float v2f;
typedef __attribute__((ext_vector_type(8))) float v8f;

// D = A(16x4 f32) * B(4x16 f32) + C(16x16 f32), full fp32 CDNA5 WMMA.
__device__ __forceinline__ v8f wmma4(v2f a, v2f b, v8f c) {
  return __builtin_amdgcn_wmma_f32_16x16x4_f32(
      /*neg_a=*/false, a, /*neg_b=*/false, b,
      /*c_mod=*/(short)0, c, /*reuse_a=*/false, /*reuse_b=*/false);
}

// ---------------------------------------------------------------------------
// C[nrows x 64] = A[nrows x 64] @ B[64 x 64 (ldb=LDB)] + bias[64]
// One block = 16 rows; wave w handles output columns [16w, 16w+16).
// LDB is compile-time so every B load folds to base + immediate offset.
// ---------------------------------------------------------------------------
template <int LDB>
__global__ void __launch_bounds__(128)
k_gemm64(const float* __restrict__ A, const float* __restrict__ Bm,
         const float* __restrict__ bias, float* __restrict__ C) {
  const int wave = threadIdx.x >> 5;
  const int lane = threadIdx.x & 31;
  const int m    = lane & 15;
  const int hi   = lane >> 4;              // 0: K+0, 1: K+2 in each A/B pair
  const int row0 = blockIdx.x * 16;
  const int n    = wave * 16 + m;

  // A operands: lane holds row m, K striped across 16 v2f (ISA 16x4 layout)
  const float* ap = A + (size_t)(row0 + m) * DD + 2 * hi;
  v2f a[16];
#pragma unroll
  for (int j = 0; j < 16; ++j) {
    a[j].x = ap[4 * j];
    a[j].y = ap[4 * j + 1];
  }

  // B base folded with lane terms; inner offsets are compile-time constants
  const float* bp = Bm + n + 2 * hi * LDB;
  v8f c = {};
#pragma unroll
  for (int j = 0; j < 16; ++j) {
    v2f b;
    b.x = bp[(4 * j) * LDB];
    b.y = bp[(4 * j + 1) * LDB];
    c = wmma4(a[j], b, c);
  }

  const float bv = bias[n];
  float* cp = C + (size_t)(row0 + 8 * hi) * DD + n;   // VGPR r -> M = r + 8*hi
#pragma unroll
  for (int r = 0; r < 8; ++r)
    cp[r * DD] = c[r] + bv;
}

// ---------------------------------------------------------------------------
// Fused FFN: Out = relu(X @ W1 + b1) @ W2 + b2   (X: nrows x 64, FF = 2048)
// Block = 16 rows, 4 waves. Each wave walks FF in 16-col chunks; the 16x16
// ReLU tile is relaid C->A layout through private LDS and immediately
// consumed by the second GEMM. 4 per-wave 16x64 partials reduced via LDS.
// ---------------------------------------------------------------------------
__global__ void __launch_bounds__(128)
k_ffn(const float* __restrict__ X,  const float* __restrict__ W1,
      const float* __restrict__ b1, const float* __restrict__ W2,
      const float* __restrict__ b2, float* __restrict__ Out) {
  __shared__ float aLds[4][16 * 16];       // per-wave relayout tile (4 KB)
  __shared__ float pLds[4][16 * 64];       // per-wave partial output (16 KB)

  const int wave = threadIdx.x >> 5;
  const int lane = threadIdx.x & 31;
  const int m    = lane & 15;
  const int hi   = lane >> 4;
  const int row0 = blockIdx.x * 16;

  // Load x tile in A layout once (K = 64)
  const float* xp = X + (size_t)(row0 + m) * DD + 2 * hi;
  v2f a0[16];
#pragma unroll
  for (int j = 0; j < 16; ++j) {
    a0[j].x = xp[4 * j];
    a0[j].y = xp[4 * j + 1];
  }

  v8f acc[4];
  {
    v8f z = {};
#pragma unroll
    for (int t = 0; t < 4; ++t) acc[t] = z;
  }

  float* myA = aLds[wave];
  // Per-chunk weight base pointers; lane terms folded once, then advanced by
  // a single 64-bit add per iteration. All inner offsets are immediates.
  const int  fc0_0 = wave * 16;
  const float* bw1 = W1 + fc0_0 + m + (size_t)(2 * hi) * FF;   // [kk][fc0+m]
  const float* bw2 = W2 + (size_t)fc0_0 * DD + m + (size_t)(2 * hi) * DD;
  const float* pb1 = b1 + fc0_0 + m;

  for (int chunk = wave; chunk < FF / 16; chunk += 4) {
    // prefetch next chunk's weights into cache (global_prefetch_b8)
    __builtin_prefetch(bw1 + 64, 0, 3);
    __builtin_prefetch(bw2 + 64 * DD, 0, 3);

    // ---- GEMM1: t(16x16) = x_tile @ W1[:, fc0:fc0+16]
    v8f t = {};
#pragma unroll
    for (int j = 0; j < 16; ++j) {
      v2f b;
      b.x = bw1[(4 * j) * FF];
      b.y = bw1[(4 * j + 1) * FF];
      t = wmma4(a0[j], b, t);
    }

    // bias + ReLU, park in LDS (row-major 16x16: M*16+N)
    const float bb = *pb1;
#pragma unroll
    for (int r = 0; r < 8; ++r) {
      float v = t[r] + bb;
      myA[(r + 8 * hi) * 16 + m] = v > 0.f ? v : 0.f;
    }

    // re-read as A operands (K = 16 for GEMM2)
    v2f a2[4];
#pragma unroll
    for (int j = 0; j < 4; ++j) {
      int k = 4 * j + 2 * hi;
      a2[j].x = myA[m * 16 + k];
      a2[j].y = myA[m * 16 + k + 1];
    }

    // ---- GEMM2: acc[ct] += relu_tile @ W2[fc0:fc0+16, 16ct:16ct+16]
#pragma unroll
    for (int ct = 0; ct < 4; ++ct) {
#pragma unroll
      for (int j = 0; j < 4; ++j) {
        v2f b;
        b.x = bw2[(4 * j) * DD + ct * 16];
        b.y = bw2[(4 * j + 1) * DD + ct * 16];
        acc[ct] = wmma4(a2[j], b, acc[ct]);
      }
    }

    bw1 += 64;            // 4 chunks ahead in the FF (column) dimension
    bw2 += 64 * DD;       // 4 chunks ahead in the FF (row) dimension
    pb1 += 64;
  }

  // dump per-wave partials, reduce across waves, add bias, store
  float* myP = pLds[wave];
#pragma unroll
  for (int ct = 0; ct < 4; ++ct)
#pragma unroll
    for (int r = 0; r < 8; ++r)
      myP[(r + 8 * hi) * 64 + ct * 16 + m] = acc[ct][r];
  __syncthreads();

  for (int idx = threadIdx.x; idx < 16 * 64; idx += 128) {
    int mm = idx >> 6, nn = idx & 63;
    float s = pLds[0][idx] + pLds[1][idx] + pLds[2][idx] + pLds[3][idx] + b2[nn];
    Out[(size_t)(row0 + mm) * DD + nn] = s;
  }
}

// ---------------------------------------------------------------------------
// degree accumulation / rsqrt / edge scatter / residual-LN / final dot
// ---------------------------------------------------------------------------
__global__ void __launch_bounds__(256)
k_deg(const int* __restrict__ row, float* __restrict__ deg, int nE) {
  int e = blockIdx.x * 256 + threadIdx.x;
  if (e < nE) unsafeAtomicAdd(&deg[row[e]], 1.0f);
}

__global__ void __launch_bounds__(256)
k_rsqrt(float* __restrict__ d, int n) {
  int i = blockIdx.x * 256 + threadIdx.x;
  if (i < n) d[i] = rsqrtf(d[i]);
}

__global__ void __launch_bounds__(256)
k_scatter(const float* __restrict__ H, const float* __restrict__ dis,
          const int* __restrict__ row, const int* __restrict__ col,
          float* __restrict__ Xout, int nE) {
  int e = blockIdx.x * 8 + (threadIdx.x >> 5);
  if (e >= nE) return;
  int lane = threadIdx.x & 31;
  int r = row[e], c = col[e];
  float nrm = dis[r] * dis[c];
  const float* hr = H + (size_t)r * DD;
  float* xo = Xout + (size_t)c * DD;
  unsafeAtomicAdd(&xo[2 * lane],     nrm * hr[2 * lane]);
  unsafeAtomicAdd(&xo[2 * lane + 1], nrm * hr[2 * lane + 1]);
}

__global__ void __launch_bounds__(256)
k_add_ln(const float* __restrict__ X, const float* __restrict__ Dlt,
         const float* __restrict__ g, const float* __restrict__ b,
         float* __restrict__ Out, int nrows) {
  int row = blockIdx.x * 8 + (threadIdx.x >> 5);
  if (row >= nrows) return;
  int lane = threadIdx.x & 31;
  const float* xr = X + (size_t)row * DD;
  const float* dr = Dlt + (size_t)row * DD;
  float f0 = xr[2 * lane] + dr[2 * lane];
  float f1 = xr[2 * lane + 1] + dr[2 * lane + 1];
  float s = f0 + f1;
  for (int off = 16; off; off >>= 1) s += __shfl_xor(s, off, 32);
  float mean = s * (1.0f / 64.0f);
  float d0 = f0 - mean, d1 = f1 - mean;
  float v = d0 * d0 + d1 * d1;
  for (int off = 16; off; off >>= 1) v += __shfl_xor(v, off, 32);
  float inv = rsqrtf(v * (1.0f / 64.0f) + 1e-5f);
  Out[(size_t)row * DD + 2 * lane]     = d0 * inv * g[2 * lane]     + b[2 * lane];
  Out[(size_t)row * DD + 2 * lane + 1] = d1 * inv * g[2 * lane + 1] + b[2 * lane + 1];
}

__global__ void __launch_bounds__(256)
k_dot(const float* __restrict__ X, const int* __restrict__ users,
      const int* __restrict__ items, float* __restrict__ out, int n) {
  int i = blockIdx.x * 8 + (threadIdx.x >> 5);
  if (i >= n) return;
  int lane = threadIdx.x & 31;
  const float* u = X + (size_t)users[i] * DD;
  const float* t = X + (size_t)(N_USERS + items[i]) * DD;
  float s = u[2 * lane] * t[2 * lane] + u[2 * lane + 1] * t[2 * lane + 1];
  for (int off = 16; off; off >>= 1) s += __shfl_xor(s, off, 32);
  if (lane == 0) out[i] = s;
}

// ---------------------------------------------------------------------------
extern "C" void kernel_launch(void* const* d_in, const int* in_sizes, int n_in,
                              void* d_out, int out_size, void* d_ws, size_t ws_size,
                              hipStream_t stream) {
  const int*   users      = (const int*)d_in[0];
  const int*   items      = (const int*)d_in[1];
  const int*   erow       = (const int*)d_in[2];
  const int*   ecol       = erow + E_TOTAL;
  const float* user_table = (const float*)d_in[3];
  const float* item_table = (const float*)d_in[4];
  const float* gl_W  = (const float*)d_in[5];
  const float* gl_b  = (const float*)d_in[6];
  const float* corr_W = (const float*)d_in[7];
  const float* corr_b = (const float*)d_in[8];
  const float* qkv_W = (const float*)d_in[9];
  const float* qkv_b = (const float*)d_in[10];
  const float* out_W = (const float*)d_in[11];
  const float* out_b = (const float*)d_in[12];
  const float* ff1_W = (const float*)d_in[13];
  const float* ff1_b = (const float*)d_in[14];
  const float* ff2_W = (const float*)d_in[15];
  const float* ff2_b = (const float*)d_in[16];
  const float* ln1_g = (const float*)d_in[17];
  const float* ln1_b = (const float*)d_in[18];
  const float* ln2_g = (const float*)d_in[19];
  const float* ln2_b = (const float*)d_in[20];

  const size_t NF = (size_t)N_NODES * DD;
  float* xA  = (float*)d_ws;
  float* xB  = xA + NF;
  float* hB  = xB + NF;
  float* dis = hB + NF;

  const int gRows = N_NODES / 16;          // 9375 (exact)
  const int gLN   = N_NODES / 8;           // 18750 (exact)
  const int gScat = (E_TOTAL + 7) / 8;

  // x0 = concat(user_table, item_table)
  hipMemcpyAsync(xA, user_table, (size_t)N_USERS * DD * 4, hipMemcpyDeviceToDevice, stream);
  hipMemcpyAsync(xA + (size_t)N_USERS * DD, item_table, (size_t)N_ITEMS * DD * 4,
                 hipMemcpyDeviceToDevice, stream);

  // degree -> dis = deg^-1/2
  hipMemsetAsync(dis, 0, (size_t)N_NODES * 4, stream);
  k_deg<<<(E_TOTAL + 255) / 256, 256, 0, stream>>>(erow, dis, E_TOTAL);
  k_rsqrt<<<(N_NODES + 255) / 256, 256, 0, stream>>>(dis, N_NODES);

  // GCN layer 0: h = xA@W0+b0 ; xB = scatter(norm*h[row] -> col)
  k_gemm64<DD><<<gRows, 128, 0, stream>>>(xA, gl_W, gl_b, hB);
  hipMemsetAsync(xB, 0, NF * 4, stream);
  k_scatter<<<gScat, 256, 0, stream>>>(hB, dis, erow, ecol, xB, E_TOTAL);

  // GCN layer 1
  k_gemm64<DD><<<gRows, 128, 0, stream>>>(xB, gl_W + DD * DD, gl_b + DD, hB);
  hipMemsetAsync(xA, 0, NF * 4, stream);
  k_scatter<<<gScat, 256, 0, stream>>>(hB, dis, erow, ecol, xA, E_TOTAL);

  // corr projection (users+items together; transformer is row-independent)
  k_gemm64<DD><<<gRows, 128, 0, stream>>>(xA, corr_W, corr_b, xB);

  // transformer layers on xB
  for (int l = 0; l < NL; ++l) {
    // v = x @ qkv_W[l][:,128:192] + qkv_b[l][128:192]
    k_gemm64<3 * DD><<<gRows, 128, 0, stream>>>(
        xB, qkv_W + (size_t)l * DD * 192 + 128, qkv_b + l * 192 + 128, hB);
    // attn = v @ out_W[l] + out_b[l]
    k_gemm64<DD><<<gRows, 128, 0, stream>>>(
        hB, out_W + (size_t)l * DD * DD, out_b + l * DD, xA);
    k_add_ln<<<gLN, 256, 0, stream>>>(xB, xA, ln1_g + l * DD, ln1_b + l * DD, xB, N_NODES);
    // fused FFN
    k_ffn<<<gRows, 128, 0, stream>>>(xB, ff1_W + (size_t)l * DD * FF, ff1_b + l * FF,
                                     ff2_W + (size_t)l * FF * DD, ff2_b + l * DD, xA);
    k_add_ln<<<gLN, 256, 0, stream>>>(xB, xA, ln2_g + l * DD, ln2_b + l * DD, xB, N_NODES);
  }

  // final pairwise dot
  k_dot<<<(B_PAIRS + 7) / 8, 256, 0, stream>>>(xB, users, items, (float*)d_out, B_PAIRS);
}